// LinearAttention_86921548137131
// MI455X (gfx1250) — compile-verified
//
#include <hip/hip_runtime.h>

typedef float v2f __attribute__((ext_vector_type(2)));
typedef float v8f __attribute__((ext_vector_type(8)));

#define NBATCH   4
#define LSEQ     8192
#define NHEAD    8
#define EDIM     64
#define NHPAIRS  (NBATCH * NHEAD)           // 32
#define SOFT_T   0.35355339059327379f       // 64^-0.25
#define EPS_F    1e-6f

#define P1_CHUNK   512
#define P1_CHUNKS  (LSEQ / P1_CHUNK)        // 16
#define SUBROWS    64                       // rows staged in LDS per iteration
#define KV_STRIDE  4160                     // 64*64 kv + 64 ksum per (n,h)
#define WS_FLOATS  (NHPAIRS * KV_STRIDE)    // 133120 floats

// elu(x*t) + 1
__device__ __forceinline__ float fmap(float x) {
    float y = x * SOFT_T;
    return (y > 0.0f) ? (y + 1.0f) : __expf(y);
}

// swizzled LDS index for K-packed-by-4 layout: [(kk>>2)][col][kk&3], col-dim = 64
__device__ __forceinline__ int swz(int kk, int col) {
    return ((kk >> 2) * 64 + col) * 4 + (kk & 3);
}

__global__ void la_zero_ws(float* __restrict__ ws) {
    int i = blockIdx.x * blockDim.x + threadIdx.x;
    if (i < WS_FLOATS) ws[i] = 0.0f;
}

// ---------------- Phase 1: kv[m,d] = sum_s f(k)[s,d]*v[s,m]; ksum[d] ----------------
__global__ __launch_bounds__(256) void la_phase1(const float* __restrict__ Kin,
                                                 const float* __restrict__ Vin,
                                                 float* __restrict__ ws) {
    __shared__ float kk[4096];   // f(k) subtile, swizzled [s][d]
    __shared__ float vv[4096];   // v subtile,    swizzled [s][m]

    const int nh    = blockIdx.x / P1_CHUNKS;
    const int chunk = blockIdx.x % P1_CHUNKS;
    const int nidx  = nh / NHEAD;
    const int hidx  = nh % NHEAD;

    const int t    = threadIdx.x;
    const int lane = t & 31;
    const int wave = t >> 5;
    const int half = lane >> 4;
    const int l16  = lane & 15;

    // two 16x16 C tiles per wave over the 4x4 tile grid
    const int tid0 = wave * 2, tid1 = tid0 + 1;
    const int mt0 = tid0 >> 2, dt0 = tid0 & 3;
    const int mt1 = tid1 >> 2, dt1 = tid1 & 3;

    v8f c0 = {}; v8f c1 = {};
    float ksacc = 0.0f;          // valid for t < 64 (d = t)

    const int base = nidx * (LSEQ * NHEAD * EDIM) + hidx * EDIM;  // + s*NHEAD*EDIM + e
    const int s0   = chunk * P1_CHUNK;

    for (int sb = 0; sb < P1_CHUNK; sb += SUBROWS) {
        // stage 64 rows x 64 cols of k (feature-mapped) and v, coalesced float4
        for (int r = 0; r < 4; ++r) {
            int idx = t + 256 * r;              // 0..1023
            int row = idx >> 4;                 // s within subtile
            int d0  = (idx & 15) << 2;          // column group
            int g   = base + (s0 + sb + row) * (NHEAD * EDIM) + d0;
            float4 kv4 = *(const float4*)(Kin + g);
            float4 vv4 = *(const float4*)(Vin + g);
            kk[swz(row, d0 + 0)] = fmap(kv4.x);
            kk[swz(row, d0 + 1)] = fmap(kv4.y);
            kk[swz(row, d0 + 2)] = fmap(kv4.z);
            kk[swz(row, d0 + 3)] = fmap(kv4.w);
            vv[swz(row, d0 + 0)] = vv4.x;
            vv[swz(row, d0 + 1)] = vv4.y;
            vv[swz(row, d0 + 2)] = vv4.z;
            vv[swz(row, d0 + 3)] = vv4.w;
        }
        __syncthreads();

        if (t < 64) {
            #pragma unroll 8
            for (int s = 0; s < SUBROWS; ++s) ksacc += kk[swz(s, t)];
        }

        // 16 K-steps of V_WMMA_F32_16X16X4_F32 per tile
        #pragma unroll 4
        for (int ks = 0; ks < SUBROWS / 4; ++ks) {
            int off = ks * 256 + 2 * half;      // K = 4*ks + 2*half + {0,1}
            v2f a0 = *(const v2f*)&vv[off + (mt0 * 16 + l16) * 4];
            v2f b0 = *(const v2f*)&kk[off + (dt0 * 16 + l16) * 4];
            c0 = __builtin_amdgcn_wmma_f32_16x16x4_f32(false, a0, false, b0,
                                                       (short)0, c0, false, false);
            v2f a1 = *(const v2f*)&vv[off + (mt1 * 16 + l16) * 4];
            v2f b1 = *(const v2f*)&kk[off + (dt1 * 16 + l16) * 4];
            c1 = __builtin_amdgcn_wmma_f32_16x16x4_f32(false, a1, false, b1,
                                                       (short)0, c1, false, false);
        }
        __syncthreads();
    }

    // combine partials: C[M=m, N=d], M = r + 8*half, N = l16
    float* kvws = ws + nh * KV_STRIDE;
    #pragma unroll 8
    for (int r = 0; r < 8; ++r) {
        int m0 = mt0 * 16 + r + 8 * half;
        int d0 = dt0 * 16 + l16;
        atomicAdd(&kvws[m0 * 64 + d0], c0[r]);
        int m1 = mt1 * 16 + r + 8 * half;
        int d1 = dt1 * 16 + l16;
        atomicAdd(&kvws[m1 * 64 + d1], c1[r]);
    }
    if (t < 64) atomicAdd(&kvws[4096 + t], ksacc);
}

// ---------------- Phase 2: out[l,m] = z[l] * sum_d f(q)[l,d]*kv[m,d] ----------------
__global__ __launch_bounds__(256) void la_phase2(const float* __restrict__ Qin,
                                                 const float* __restrict__ ws,
                                                 float* __restrict__ out) {
    __shared__ float qq[4096];   // f(q) tile, swizzled [d][l]
    __shared__ float kb[4096];   // kv^T,      swizzled [d][m]
    __shared__ float kss[64];
    __shared__ float zz[64];

    const int nh = blockIdx.x >> 7;          // / 128
    const int lt = blockIdx.x & 127;         // 64-row tile index
    const int nidx = nh / NHEAD;
    const int hidx = nh % NHEAD;

    const int t    = threadIdx.x;
    const int lane = t & 31;
    const int wave = t >> 5;
    const int half = lane >> 4;
    const int l16  = lane & 15;

    const float* kvws = ws + nh * KV_STRIDE;
    const int base = nidx * (LSEQ * NHEAD * EDIM) + hidx * EDIM;
    const int l0   = lt * 64;

    // stage kv (transpose into swizzle) and f(q) tile
    for (int r = 0; r < 4; ++r) {
        int idx = t + 256 * r;
        int row = idx >> 4;                  // m for kv, l-local for q
        int d0  = (idx & 15) << 2;
        float4 kv4 = *(const float4*)(kvws + row * 64 + d0);
        kb[swz(d0 + 0, row)] = kv4.x;
        kb[swz(d0 + 1, row)] = kv4.y;
        kb[swz(d0 + 2, row)] = kv4.z;
        kb[swz(d0 + 3, row)] = kv4.w;
        float4 q4 = *(const float4*)(Qin + base + (l0 + row) * (NHEAD * EDIM) + d0);
        qq[swz(d0 + 0, row)] = fmap(q4.x);
        qq[swz(d0 + 1, row)] = fmap(q4.y);
        qq[swz(d0 + 2, row)] = fmap(q4.z);
        qq[swz(d0 + 3, row)] = fmap(q4.w);
    }
    if (t < 64) kss[t] = kvws[4096 + t];
    __syncthreads();

    if (t < 64) {
        float acc = 0.0f;
        #pragma unroll 8
        for (int d = 0; d < 64; ++d) acc += qq[swz(d, t)] * kss[d];
        zz[t] = 1.0f / (acc + EPS_F);
    }
    __syncthreads();

    const int tid0 = wave * 2, tid1 = tid0 + 1;
    const int lt0 = tid0 >> 2, mt0 = tid0 & 3;
    const int lt1 = tid1 >> 2, mt1 = tid1 & 3;

    v8f c0 = {}; v8f c1 = {};
    #pragma unroll 4
    for (int ks = 0; ks < 16; ++ks) {        // K = d
        int off = ks * 256 + 2 * half;
        v2f a0 = *(const v2f*)&qq[off + (lt0 * 16 + l16) * 4];
        v2f b0 = *(const v2f*)&kb[off + (mt0 * 16 + l16) * 4];
        c0 = __builtin_amdgcn_wmma_f32_16x16x4_f32(false, a0, false, b0,
                                                   (short)0, c0, false, false);
        v2f a1 = *(const v2f*)&qq[off + (lt1 * 16 + l16) * 4];
        v2f b1 = *(const v2f*)&kb[off + (mt1 * 16 + l16) * 4];
        c1 = __builtin_amdgcn_wmma_f32_16x16x4_f32(false, a1, false, b1,
                                                   (short)0, c1, false, false);
    }

    // scale by z and store: D[M=l_local, N=m], M = r + 8*half, N = l16
    #pragma unroll 8
    for (int r = 0; r < 8; ++r) {
        int ll0 = lt0 * 16 + r + 8 * half;
        out[base + (l0 + ll0) * (NHEAD * EDIM) + mt0 * 16 + l16] = c0[r] * zz[ll0];
        int ll1 = lt1 * 16 + r + 8 * half;
        out[base + (l0 + ll1) * (NHEAD * EDIM) + mt1 * 16 + l16] = c1[r] * zz[ll1];
    }
}

extern "C" void kernel_launch(void* const* d_in, const int* in_sizes, int n_in,
                              void* d_out, int out_size, void* d_ws, size_t ws_size,
                              hipStream_t stream) {
    const float* q = (const float*)d_in[0];
    const float* k = (const float*)d_in[1];
    const float* v = (const float*)d_in[2];
    float* outp = (float*)d_out;
    float* ws   = (float*)d_ws;
    (void)in_sizes; (void)n_in; (void)out_size; (void)ws_size;

    la_zero_ws<<<(WS_FLOATS + 255) / 256, 256, 0, stream>>>(ws);
    la_phase1<<<NHPAIRS * P1_CHUNKS, 256, 0, stream>>>(k, v, ws);
    la_phase2<<<NHPAIRS * (LSEQ / 64), 256, 0, stream>>>(q, ws, outp);
}